// DiversityLoss_65343632441392
// MI455X (gfx1250) — compile-verified
//
#include <hip/hip_runtime.h>

// Diversity loss (cosine path), algebraically reduced:
//   sum_{t,s} xn[t].xn[s]  ==  || sum_t xn[t] ||^2
// => one streaming pass over target (8 MB), HBM-bound (~0.35us @ 23.3 TB/s).
// The weighted column-sum v = w^T X uses V_WMMA_F32_16X16X4_F32 (full fp32):
//   A (16x4) = transposed 16-d x 4-row tile of X (coalesced lane-varying-d loads)
//   B (4x16) = the 4 row weights broadcast across N
//   C accumulates v[d] (N-redundant; compute is free at this arithmetic intensity).
// Row-norm butterfly uses DPP16 row_xmask (VALU-only, no LDS/dscnt waits).

typedef __attribute__((ext_vector_type(2))) float v2f;
typedef __attribute__((ext_vector_type(8))) float v8f;

#define EPS 1e-8f
#define SPLIT 16     // T-splits per batch
#define WAVES 16     // waves per block
#define TPB (WAVES * 32)

// Sum across each 16-lane half via DPP row_xmask butterfly (xor 1,2,4,8).
__device__ __forceinline__ float xor_add16(float v) {
  int x;
  x = __builtin_amdgcn_update_dpp(0, __float_as_int(v), 0x161, 0xf, 0xf, true);
  v += __int_as_float(x);
  x = __builtin_amdgcn_update_dpp(0, __float_as_int(v), 0x162, 0xf, 0xf, true);
  v += __int_as_float(x);
  x = __builtin_amdgcn_update_dpp(0, __float_as_int(v), 0x164, 0xf, 0xf, true);
  v += __int_as_float(x);
  x = __builtin_amdgcn_update_dpp(0, __float_as_int(v), 0x168, 0xf, 0xf, true);
  v += __int_as_float(x);
  return v;
}

__global__ __launch_bounds__(TPB) void div_partial_kernel(
    const float* __restrict__ X, const int* __restrict__ lens,
    float* __restrict__ part, int T) {
  __shared__ float vsh[64];
  const int b    = blockIdx.y;
  const int s    = blockIdx.x;
  const int tid  = (int)threadIdx.x;
  const int lane = tid & 31;
  const int wave = tid >> 5;
  const int half = lane >> 4;   // 0: lanes 0-15 supply K=0,1   1: lanes 16-31 supply K=2,3
  const int lm   = lane & 15;   // M index (d within a 16-chunk)
  const int len  = lens[b];
  const float* Xb = X + (size_t)b * (size_t)T * 64;

  if (tid < 64) vsh[tid] = 0.0f;
  __syncthreads();

  const int rows_per_block = (T + SPLIT - 1) / SPLIT;
  const int row0 = s * rows_per_block;
  int row1 = row0 + rows_per_block;
  if (row1 > T) row1 = T;
  const int nrows = row1 - row0;
  const int nfull = nrows >> 2;              // fully-populated 4-row groups

  v8f acc0 = {0,0,0,0,0,0,0,0};
  v8f acc1 = acc0, acc2 = acc0, acc3 = acc0;

  // ---- fast path: no per-lane guards, pointer-increment addressing ----
  int t = row0 + wave * 4 + half * 2;                          // lane's first row
  const float* rA = Xb + (size_t)(row0 + wave * 4 + half * 2) * 64 + lm;

  for (int g = wave; g < nfull; g += WAVES) {
    const float* rB = rA + 64;
    const float ax0 = rA[0], ax1 = rA[16], ax2 = rA[32], ax3 = rA[48];
    const float ay0 = rB[0], ay1 = rB[16], ay2 = rB[32], ay3 = rB[48];

    // Row sum-of-squares: DPP butterfly within each 16-lane half.
    float sx = xor_add16(ax0*ax0 + ax1*ax1 + ax2*ax2 + ax3*ax3);
    float sy = xor_add16(ay0*ay0 + ay1*ay1 + ay2*ay2 + ay3*ay3);

    const float wA = (t     < len) ? (1.0f / fmaxf(sqrtf(sx), EPS)) : 0.0f;
    const float wB = (t + 1 < len) ? (1.0f / fmaxf(sqrtf(sy), EPS)) : 0.0f;

    v2f bv; bv.x = wA;  bv.y = wB;
    v2f a0; a0.x = ax0; a0.y = ay0;
    v2f a1; a1.x = ax1; a1.y = ay1;
    v2f a2; a2.x = ax2; a2.y = ay2;
    v2f a3; a3.x = ax3; a3.y = ay3;

    acc0 = __builtin_amdgcn_wmma_f32_16x16x4_f32(false, a0, false, bv, (short)0, acc0, false, false);
    acc1 = __builtin_amdgcn_wmma_f32_16x16x4_f32(false, a1, false, bv, (short)0, acc1, false, false);
    acc2 = __builtin_amdgcn_wmma_f32_16x16x4_f32(false, a2, false, bv, (short)0, acc2, false, false);
    acc3 = __builtin_amdgcn_wmma_f32_16x16x4_f32(false, a3, false, bv, (short)0, acc3, false, false);

    rA += (size_t)WAVES * 4 * 64;
    t  += WAVES * 4;
  }

  // ---- tail group (nrows % 4 != 0): one wave, guarded loads, EXEC full at WMMA ----
  if ((nrows & 3) && wave == (nfull % WAVES)) {
    const int tbase = row0 + nfull * 4;
    const int tA = tbase + half * 2;
    const int tB = tA + 1;
    float ax0=0.f, ax1=0.f, ax2=0.f, ax3=0.f;
    float ay0=0.f, ay1=0.f, ay2=0.f, ay3=0.f;
    if (tA < row1) {
      const float* r = Xb + (size_t)tA * 64 + lm;
      ax0 = r[0]; ax1 = r[16]; ax2 = r[32]; ax3 = r[48];
    }
    if (tB < row1) {
      const float* r = Xb + (size_t)tB * 64 + lm;
      ay0 = r[0]; ay1 = r[16]; ay2 = r[32]; ay3 = r[48];
    }
    float sx = xor_add16(ax0*ax0 + ax1*ax1 + ax2*ax2 + ax3*ax3);
    float sy = xor_add16(ay0*ay0 + ay1*ay1 + ay2*ay2 + ay3*ay3);
    const float wA = (tA < row1 && tA < len) ? (1.0f / fmaxf(sqrtf(sx), EPS)) : 0.0f;
    const float wB = (tB < row1 && tB < len) ? (1.0f / fmaxf(sqrtf(sy), EPS)) : 0.0f;

    v2f bv; bv.x = wA;  bv.y = wB;
    v2f a0; a0.x = ax0; a0.y = ay0;
    v2f a1; a1.x = ax1; a1.y = ay1;
    v2f a2; a2.x = ax2; a2.y = ay2;
    v2f a3; a3.x = ax3; a3.y = ay3;

    acc0 = __builtin_amdgcn_wmma_f32_16x16x4_f32(false, a0, false, bv, (short)0, acc0, false, false);
    acc1 = __builtin_amdgcn_wmma_f32_16x16x4_f32(false, a1, false, bv, (short)0, acc1, false, false);
    acc2 = __builtin_amdgcn_wmma_f32_16x16x4_f32(false, a2, false, bv, (short)0, acc2, false, false);
    acc3 = __builtin_amdgcn_wmma_f32_16x16x4_f32(false, a3, false, bv, (short)0, acc3, false, false);
  }

  // Combine waves: C layout => VGPR j holds d = dbase + half*8 + j (same on all n lanes).
  if (lm == 0) {
#pragma unroll
    for (int j = 0; j < 8; ++j) {
      atomicAdd(&vsh[ 0 + half*8 + j], acc0[j]);
      atomicAdd(&vsh[16 + half*8 + j], acc1[j]);
      atomicAdd(&vsh[32 + half*8 + j], acc2[j]);
      atomicAdd(&vsh[48 + half*8 + j], acc3[j]);
    }
  }
  __syncthreads();

  if (tid < 64) part[((size_t)b * SPLIT + s) * 64 + tid] = vsh[tid];
}

__global__ __launch_bounds__(32) void div_final_kernel(
    const float* __restrict__ part, const int* __restrict__ lens,
    float* __restrict__ out, int B) {
  const int lane = (int)threadIdx.x;
  float score = 0.0f;
  int cnt = 0;
  for (int b = 0; b < B; ++b) {
    float v0 = 0.0f, v1 = 0.0f;
    for (int s = 0; s < SPLIT; ++s) {          // fixed-order, deterministic sum
      const float* p = part + ((size_t)b * SPLIT + s) * 64;
      v0 += p[lane];
      v1 += p[lane + 32];
    }
    float ss = v0 * v0 + v1 * v1;              // ||v||^2 partial
#pragma unroll
    for (int off = 1; off < 32; off <<= 1) ss += __shfl_xor(ss, off, 32);
    const int len = lens[b];
    if (len > 1) {
      const float lf = (float)len;
      score += (ss - lf) / (lf * (lf - 1.0f));
    }
    if (len != 1) cnt++;
  }
  if (lane == 0) out[0] = score / (float)cnt;
}

extern "C" void kernel_launch(void* const* d_in, const int* in_sizes, int n_in,
                              void* d_out, int out_size, void* d_ws, size_t ws_size,
                              hipStream_t stream) {
  const float* X   = (const float*)d_in[0];   // [B, T, 64] f32
  const int* lens  = (const int*)d_in[1];     // [B] i32
  const int B = in_sizes[1];
  const int T = in_sizes[0] / (B * 64);       // D = 64 per reference
  float* part = (float*)d_ws;                 // B*SPLIT*64 floats (32 KB for B=8)

  dim3 grid(SPLIT, B);
  div_partial_kernel<<<grid, TPB, 0, stream>>>(X, lens, part, T);
  div_final_kernel<<<1, 32, 0, stream>>>(part, lens, (float*)d_out, B);
}